// ExpressionModule_2267742732789
// MI455X (gfx1250) — compile-verified
//
#include <hip/hip_runtime.h>

// Element-wise expression tree over 16M fp32 (bandwidth-bound streaming map):
//   s = x*x ; t_i = tanh(g_i*s) x8 ; u_j = w0*t + w1*t' + b x4 ;
//   v_k = tanh(g*u*u') x2 ; out = w0*v0 + w1*v1 + b
// Roofline: 64 MB in + 64 MB out, touched once -> ~5.5 us floor @ 23.3 TB/s.
// CDNA5 levers: v_tanh_f32 (1 TRANS/tanh, co-executes with VALU),
// v_pk_fma_f32 / v_pk_mul_f32 (2x f32 per VALU issue) for all non-tanh math,
// B128 non-temporal loads/stores, params in SGPRs via s_load_b512/b256.

typedef __attribute__((ext_vector_type(4))) float v4f;
typedef __attribute__((ext_vector_type(2))) float v2f;

__device__ __forceinline__ float fast_tanh(float v) {
#if __has_builtin(__builtin_amdgcn_tanhf)
    return __builtin_amdgcn_tanhf(v);   // v_tanh_f32 (gfx1250 transcendental)
#else
    float e = __builtin_amdgcn_exp2f(2.8853900817779268f * v); // 2*log2(e)
    return 1.0f - 2.0f * __builtin_amdgcn_rcpf(e + 1.0f);
#endif
}

__device__ __forceinline__ v2f tanh2(v2f v) {
    v2f r;
    r.x = fast_tanh(v.x);
    r.y = fast_tanh(v.y);
    return r;
}

struct P25 { float a[25]; };

// Preorder param indices:
//  root waff: 0,1,2
//   L gmul: 3
//     LL waff: 4,5,6    -> leaf gmuls 7, 8
//     LR waff: 9,10,11  -> leaf gmuls 12, 13
//   R gmul: 14
//     RL waff: 15,16,17 -> leaf gmuls 18, 19
//     RR waff: 20,21,22 -> leaf gmuls 23, 24

// Two elements at once: all non-tanh math in packed fp32 (v_pk_*).
__device__ __forceinline__ v2f eval2(v2f x, const P25& q) {
    v2f s  = x * x;                                   // v_pk_mul_f32
    v2f t0 = tanh2(q.a[7]  * s);
    v2f t1 = tanh2(q.a[8]  * s);
    v2f t2 = tanh2(q.a[12] * s);
    v2f t3 = tanh2(q.a[13] * s);
    v2f t4 = tanh2(q.a[18] * s);
    v2f t5 = tanh2(q.a[19] * s);
    v2f t6 = tanh2(q.a[23] * s);
    v2f t7 = tanh2(q.a[24] * s);
    // waffs: two v_pk_fma_f32 each
    v2f u0 = __builtin_elementwise_fma((v2f)q.a[4],  t0,
             __builtin_elementwise_fma((v2f)q.a[5],  t1, (v2f)q.a[6]));
    v2f u1 = __builtin_elementwise_fma((v2f)q.a[9],  t2,
             __builtin_elementwise_fma((v2f)q.a[10], t3, (v2f)q.a[11]));
    v2f u2 = __builtin_elementwise_fma((v2f)q.a[15], t4,
             __builtin_elementwise_fma((v2f)q.a[16], t5, (v2f)q.a[17]));
    v2f u3 = __builtin_elementwise_fma((v2f)q.a[20], t6,
             __builtin_elementwise_fma((v2f)q.a[21], t7, (v2f)q.a[22]));
    v2f v0 = tanh2(q.a[3]  * u0 * u1);                // 2x v_pk_mul_f32 + tanh
    v2f v1 = tanh2(q.a[14] * u2 * u3);
    return __builtin_elementwise_fma((v2f)q.a[0], v0,
           __builtin_elementwise_fma((v2f)q.a[1], v1, (v2f)q.a[2]));
}

// Hot kernel: pure vectorized stream, no tail code.
__global__ void __launch_bounds__(256)
expr_tree_kernel(const float* __restrict__ x,
                 const float* __restrict__ params,
                 float* __restrict__ out,
                 int nvec)   // number of float4 chunks
{
    P25 q;
#pragma unroll
    for (int i = 0; i < 25; ++i) q.a[i] = params[i];

    const v4f* __restrict__ xv = (const v4f*)x;
    v4f* __restrict__ ov       = (v4f*)out;

    int tid    = blockIdx.x * blockDim.x + threadIdx.x;
    int stride = gridDim.x * blockDim.x;

    for (int i = tid; i < nvec; i += stride) {
        v4f xin = __builtin_nontemporal_load(&xv[i]);
        v2f lo = { xin.x, xin.y };
        v2f hi = { xin.z, xin.w };
        v2f rlo = eval2(lo, q);
        v2f rhi = eval2(hi, q);
        v4f r = { rlo.x, rlo.y, rhi.x, rhi.y };
        __builtin_nontemporal_store(r, &ov[i]);
    }
}

// Tail kernel: scalar path, launched only if N % 4 != 0 (never for N = 16M).
__global__ void __launch_bounds__(64)
expr_tree_tail(const float* __restrict__ x,
               const float* __restrict__ params,
               float* __restrict__ out,
               int base, int rem)
{
    P25 q;
#pragma unroll
    for (int i = 0; i < 25; ++i) q.a[i] = params[i];

    int t = threadIdx.x;
    if (t < rem) {
        int idx = base + t;
        v2f xin = { x[idx], 0.0f };
        out[idx] = eval2(xin, q).x;
    }
}

extern "C" void kernel_launch(void* const* d_in, const int* in_sizes, int n_in,
                              void* d_out, int out_size, void* d_ws, size_t ws_size,
                              hipStream_t stream) {
    const float* x      = (const float*)d_in[0];
    const float* params = (const float*)d_in[1];
    float* out          = (float*)d_out;

    int n    = in_sizes[0];          // 16777216
    int nvec = n >> 2;               // float4 chunks
    int rem  = n - (nvec << 2);      // 0 for N = 16M

    const int threads = 256;
    int blocks = (nvec + threads - 1) / threads;
    if (blocks < 1) blocks = 1;

    if (nvec > 0)
        expr_tree_kernel<<<blocks, threads, 0, stream>>>(x, params, out, nvec);
    if (rem > 0)  // deterministic: depends only on input size
        expr_tree_tail<<<1, 64, 0, stream>>>(x, params, out, nvec << 2, rem);
}